// GATLayer_24034636989186
// MI455X (gfx1250) — compile-verified
//
#include <hip/hip_runtime.h>
#include <stdint.h>

typedef __attribute__((ext_vector_type(2))) float v2f;
typedef __attribute__((ext_vector_type(8))) float v8f;

#define BQ   4
#define SQ   256
#define NN   257      // nodes
#define EE   256      // edges
#define DIN  128
#define DOUT 128
#define EDIM 16
#define MPAD 272      // 17*16, padded node rows for WMMA tiles
#define MT   17
#define NVEC ((NN*DOUT)/4)   // 8224 float4 elements in node buffers

// One workgroup per (b,s): GEMM h=x@W via f32 WMMA into LDS, then attention
// logits, masked softmax, LDS-atomic scatter-add, fused elu + store.
__global__ __launch_bounds__(256)
void gat_fused_kernel(const float* __restrict__ x,
                      const int* __restrict__ edge_index,
                      const float* __restrict__ edge_attr,
                      const uint8_t* __restrict__ edge_mask,
                      const float* __restrict__ W,
                      const float* __restrict__ avec,
                      float* __restrict__ out)
{
    __shared__ float hs[MPAD * DOUT];     // 139,264 B : h = x@W (padded rows)
    __shared__ float agg[NN * DOUT];      // 131,584 B : scatter accumulator
    __shared__ float e_lds[EE];
    __shared__ float alpha_lds[EE];
    __shared__ float red[16];

    const int bs   = blockIdx.x;          // (b,s) flat
    const int tid  = threadIdx.x;
    const int lane = tid & 31;
    const int wid  = tid >> 5;            // 8 waves
    const int half = lane >> 4;           // 0: lanes 0-15, 1: lanes 16-31
    const int l16  = lane & 15;

    const float*   xb  = x          + (size_t)bs * NN * DIN;
    const int*     eib = edge_index + (size_t)bs * EE * 2;
    const float*   eab = edge_attr  + (size_t)bs * EE * EDIM;
    const uint8_t* emb = edge_mask  + (size_t)bs * EE;

    // ---- zero the aggregation buffer (b128 stores) ----
    {
        float4 z = make_float4(0.f, 0.f, 0.f, 0.f);
        float4* agg4 = (float4*)agg;
        for (int i = tid; i < NVEC; i += 256) agg4[i] = z;
    }

    // ================= GEMM: hs = x @ W  (M=257 pad 272, K=128, N=128) =========
    // Each wave owns one 16-wide N column block (nt == wid) for all 17 M tiles,
    // so its 32 B fragments are loop-invariant and live in registers.
    const int nt   = wid;
    const int ncol = nt * 16 + l16;

    v2f bfrag[32];
    #pragma unroll
    for (int kk = 0; kk < 32; ++kk) {
        const int kb = kk * 4 + 2 * half;
        bfrag[kk].x = W[kb * DOUT + ncol];
        bfrag[kk].y = W[(kb + 1) * DOUT + ncol];
    }

    // full tiles mt = 0..15 : rows <= 255, always in bounds -> branchless loads
    for (int mt = 0; mt < MT - 1; ++mt) {
        const int mrow = mt * 16 + l16;
        const float* arow = xb + mrow * DIN + 2 * half;
        v2f af[32];
        #pragma unroll
        for (int kk = 0; kk < 32; ++kk)
            af[kk] = *(const v2f*)(arow + kk * 4);     // 32 global_load_b64 in flight
        v8f c = {};
        #pragma unroll
        for (int kk = 0; kk < 32; ++kk)
            c = __builtin_amdgcn_wmma_f32_16x16x4_f32(false, af[kk], false, bfrag[kk],
                                                      (short)0, c, false, false);
        #pragma unroll
        for (int r = 0; r < 8; ++r)
            hs[(mt * 16 + r + half * 8) * DOUT + ncol] = c[r];
    }

    // tail tile mt = 16 : only row 256 valid -> clamped address + 0/1 scale
    {
        const int   mrow = 256 + l16;
        const int   mcl  = (mrow < NN) ? mrow : 0;
        const float vsc  = (mrow < NN) ? 1.0f : 0.0f;
        const float* arow = xb + mcl * DIN + 2 * half;
        v8f c = {};
        #pragma unroll
        for (int kk = 0; kk < 32; ++kk) {
            v2f a = *(const v2f*)(arow + kk * 4);
            a.x *= vsc;  a.y *= vsc;
            c = __builtin_amdgcn_wmma_f32_16x16x4_f32(false, a, false, bfrag[kk],
                                                      (short)0, c, false, false);
        }
        #pragma unroll
        for (int r = 0; r < 8; ++r)
            hs[(256 + r + half * 8) * DOUT + ncol] = c[r];
    }
    __syncthreads();

    // ================= attention logits: one wave per edge ====================
    // lane k-slices (stride 32 -> bank-conflict-free); coefficients hoisted.
    float aws[4], awt[4];
    #pragma unroll
    for (int j = 0; j < 4; ++j) {
        aws[j] = avec[lane + 32 * j];
        awt[j] = avec[DOUT + lane + 32 * j];
    }
    const float aedge = (lane < EDIM) ? avec[2 * DOUT + lane] : 0.0f;

    for (int i = 0; i < 32; ++i) {
        const int e = wid * 32 + i;
        const int2 ei2 = *(const int2*)(eib + e * 2);   // one b64 load
        const int sIdx = (ei2.x >= 0) ? ei2.x : 0;
        const int tIdx = (ei2.y >= 0) ? ei2.y : 0;
        const float sOk = (ei2.x >= 0) ? 1.0f : 0.0f;
        const float tOk = (ei2.y >= 0) ? 1.0f : 0.0f;
        float ps = 0.0f, pt = 0.0f;
        #pragma unroll
        for (int j = 0; j < 4; ++j) {
            const int k = lane + 32 * j;
            ps += hs[sIdx * DOUT + k] * aws[j];
            pt += hs[tIdx * DOUT + k] * awt[j];
        }
        float p = sOk * ps + tOk * pt
                + eab[e * EDIM + (lane & 15)] * aedge;  // aedge==0 for lanes>=16
        #pragma unroll
        for (int off = 16; off >= 1; off >>= 1) p += __shfl_xor(p, off, 32);
        if (lane == 0) e_lds[e] = (p > 0.0f) ? p : 0.2f * p;   // leaky_relu(0.2)
    }
    __syncthreads();

    // ================= masked softmax over 256 edges ==========================
    const float masked = emb[tid] ? e_lds[tid] : -INFINITY;
    float v = masked;
    #pragma unroll
    for (int off = 16; off >= 1; off >>= 1) v = fmaxf(v, __shfl_xor(v, off, 32));
    if (lane == 0) red[wid] = v;
    __syncthreads();
    if (tid == 0) {
        float m = red[0];
        for (int i = 1; i < 8; ++i) m = fmaxf(m, red[i]);
        red[8] = m;
    }
    __syncthreads();
    const float mx = red[8];
    // all edges masked -> reference zeroes logits -> uniform softmax
    const float ex = (mx == -INFINITY) ? 1.0f
                   : ((masked == -INFINITY) ? 0.0f : __expf(masked - mx));
    v = ex;
    #pragma unroll
    for (int off = 16; off >= 1; off >>= 1) v += __shfl_xor(v, off, 32);
    if (lane == 0) red[wid] = v;
    __syncthreads();
    if (tid == 0) {
        float s = 0.0f;
        for (int i = 0; i < 8; ++i) s += red[i];
        red[9] = s;
    }
    __syncthreads();
    alpha_lds[tid] = ex / red[9];
    __syncthreads();

    // ====== scatter-add: agg[vtgt] += h[vsrc]*alpha ; agg[vsrc] += h[vtgt]*alpha
    for (int i = 0; i < 32; ++i) {
        const int e = wid * 32 + i;
        const int2 ei2 = *(const int2*)(eib + e * 2);
        const bool m = emb[e] != 0;
        const int vs = m ? ei2.x : 0;
        const int vt = m ? ei2.y : 0;
        const int gs = (vs < 0) ? 0 : vs;     // gather clamps (JAX take_along_axis)
        const int gt = (vt < 0) ? 0 : vt;
        const float al = alpha_lds[e];
        #pragma unroll
        for (int j = 0; j < 4; ++j) {
            const int k = lane + 32 * j;      // stride 32 of 64 banks: conflict-free
            if (vt >= 0) atomicAdd(&agg[vt * DOUT + k], hs[gs * DOUT + k] * al);
            if (vs >= 0) atomicAdd(&agg[vs * DOUT + k], hs[gt * DOUT + k] * al);
        }
    }
    __syncthreads();

    // ================= elu + store (b128) =====================================
    {
        float4* ob4 = (float4*)(out + (size_t)bs * NN * DOUT);
        const float4* agg4 = (const float4*)agg;
        for (int i = tid; i < NVEC; i += 256) {
            float4 va = agg4[i];
            va.x = (va.x > 0.0f) ? va.x : (__expf(va.x) - 1.0f);
            va.y = (va.y > 0.0f) ? va.y : (__expf(va.y) - 1.0f);
            va.z = (va.z > 0.0f) ? va.z : (__expf(va.z) - 1.0f);
            va.w = (va.w > 0.0f) ? va.w : (__expf(va.w) - 1.0f);
            ob4[i] = va;
        }
    }
}

extern "C" void kernel_launch(void* const* d_in, const int* in_sizes, int n_in,
                              void* d_out, int out_size, void* d_ws, size_t ws_size,
                              hipStream_t stream) {
    const float*   x  = (const float*)d_in[0];
    const int*     ei = (const int*)d_in[1];
    const float*   ea = (const float*)d_in[2];
    // d_in[3] = node_mask (unused by the reference forward)
    const uint8_t* em = (const uint8_t*)d_in[4];
    const float*   W  = (const float*)d_in[5];
    const float*   a  = (const float*)d_in[6];
    float* out = (float*)d_out;
    gat_fused_kernel<<<BQ * SQ, 256, 0, stream>>>(x, ei, ea, em, W, a, out);
}